// GATLayer_67035849556346
// MI455X (gfx1250) — compile-verified
//
#include <hip/hip_runtime.h>

typedef __attribute__((ext_vector_type(2))) float v2f;
typedef __attribute__((ext_vector_type(8))) float v8f;

#define INC   128   // input channels
#define OUTC  128   // H*C
#define HEADS 4
#define CPH   32
#define NEG_SLOPE 0.2f
#define BN_EPS 1e-5f

// ---------------------------------------------------------------------------
// 0) workspace init (graph-capture safe; no memset)
// ---------------------------------------------------------------------------
__global__ void init_ws_kernel(float* __restrict__ acc, unsigned* __restrict__ mord,
                               float* __restrict__ denom, float* __restrict__ sums,
                               float* __restrict__ sumsq, int n) {
    int t = blockIdx.x * blockDim.x + threadIdx.x;
    int stride = gridDim.x * blockDim.x;
    int total = n * OUTC;
    for (int i = t; i < total; i += stride) acc[i] = 0.0f;
    int nh = n * HEADS;
    for (int i = t; i < nh; i += stride) { mord[i] = 0x007FFFFFu; /* ord(-inf) */ denom[i] = 0.0f; }
    if (t < OUTC) { sums[t] = 0.0f; sumsq[t] = 0.0f; }
}

// ---------------------------------------------------------------------------
// 1) h = x @ W via V_WMMA_F32_16X16X4_F32.
//    One wave per 16x16 output tile; block = 8 waves = one 16-row stripe,
//    each wave owns one of the 8 column tiles (shares x rows via WGP$).
//    A layout (16x4 f32):  v0 = K0 (lanes 0-15) / K2 (lanes 16-31), v1 = K1/K3
//    B layout (4x16 f32):  v0 = row K0 / K2, v1 = row K1 / K3 (N = lane%16)
//    C/D layout (16x16):   vgpr i -> row i (lanes 0-15), row i+8 (lanes 16-31)
// ---------------------------------------------------------------------------
__global__ void __launch_bounds__(256) gemm_xw_wmma(const float* __restrict__ x,
                                                    const float* __restrict__ W,
                                                    float* __restrict__ h) {
    const int lane    = threadIdx.x & 31;
    const int wave    = threadIdx.x >> 5;      // 0..7  -> column tile
    const int rowBase = blockIdx.x << 4;       // 16 rows per block
    const int colBase = wave << 4;
    const int l       = lane & 15;
    const int half    = lane >> 4;

    const float* xrow = x + (size_t)(rowBase + l) * INC;
    v8f acc = {};
#pragma unroll
    for (int k0 = 0; k0 < INC; k0 += 4) {
        v2f a = *(const v2f*)(xrow + k0 + 2 * half);           // 8B aligned
        v2f b;
        b.x = W[(size_t)(k0 + 2 * half)     * OUTC + colBase + l];
        b.y = W[(size_t)(k0 + 2 * half + 1) * OUTC + colBase + l];
        acc = __builtin_amdgcn_wmma_f32_16x16x4_f32(
            /*neg_a=*/false, a, /*neg_b=*/false, b,
            /*c_mod=*/(short)0, acc, /*reuse_a=*/false, /*reuse_b=*/false);
    }
#pragma unroll
    for (int i = 0; i < 8; ++i)
        h[(size_t)(rowBase + i + 8 * half) * OUTC + colBase + l] = acc[i];
}

// ---------------------------------------------------------------------------
// 2) a_src[n,hd] = <h[n,hd,:], att_src[hd,:]>, same for a_dst
// ---------------------------------------------------------------------------
__global__ void attn_coef_kernel(const float* __restrict__ h,
                                 const float* __restrict__ att_src,
                                 const float* __restrict__ att_dst,
                                 float* __restrict__ as, float* __restrict__ ad, int n) {
    int t = blockIdx.x * blockDim.x + threadIdx.x;
    if (t >= n * HEADS) return;
    int node = t >> 2, hd = t & 3;
    const float* hp = h + (size_t)node * OUTC + hd * CPH;
    const float* sp = att_src + hd * CPH;
    const float* dp = att_dst + hd * CPH;
    float s = 0.0f, d = 0.0f;
#pragma unroll
    for (int c = 0; c < CPH; ++c) { float v = hp[c]; s += v * sp[c]; d += v * dp[c]; }
    as[t] = s; ad[t] = d;
}

// monotonic float -> uint encoding (total order, handles negatives)
__device__ __forceinline__ unsigned f2ord(float f) {
    unsigned u = __float_as_uint(f);
    return (u & 0x80000000u) ? ~u : (u | 0x80000000u);
}
__device__ __forceinline__ float ord2f(unsigned u) {
    u = (u & 0x80000000u) ? (u & 0x7FFFFFFFu) : ~u;
    return __uint_as_float(u);
}

__device__ __forceinline__ void edge_sd(const long long* ei, int idx, int E_, int& s, int& d) {
    if (idx < E_) { s = (int)ei[idx]; d = (int)ei[E_ + idx]; }
    else          { s = d = idx - E_; }   // self loop
}

// ---------------------------------------------------------------------------
// 3) segment max of leaky_relu(a_src[s]+a_dst[d]) over dst
// ---------------------------------------------------------------------------
__global__ void edge_max_kernel(const long long* __restrict__ ei,
                                const float* __restrict__ as, const float* __restrict__ ad,
                                unsigned* __restrict__ mord, int E_, int ET) {
    int t = blockIdx.x * blockDim.x + threadIdx.x;
    if (t >= ET * HEADS) return;
    int idx = t >> 2, hd = t & 3;
    int s, d; edge_sd(ei, idx, E_, s, d);
    float e = as[s * HEADS + hd] + ad[d * HEADS + hd];
    e = (e >= 0.0f) ? e : NEG_SLOPE * e;
    atomicMax(&mord[d * HEADS + hd], f2ord(e));
}

// ---------------------------------------------------------------------------
// 4) denom = segment_sum(exp(e - m[dst]))
// ---------------------------------------------------------------------------
__global__ void edge_sum_kernel(const long long* __restrict__ ei,
                                const float* __restrict__ as, const float* __restrict__ ad,
                                const unsigned* __restrict__ mord,
                                float* __restrict__ denom, int E_, int ET) {
    int t = blockIdx.x * blockDim.x + threadIdx.x;
    if (t >= ET * HEADS) return;
    int idx = t >> 2, hd = t & 3;
    int s, d; edge_sd(ei, idx, E_, s, d);
    float e = as[s * HEADS + hd] + ad[d * HEADS + hd];
    e = (e >= 0.0f) ? e : NEG_SLOPE * e;
    float ex = expf(e - ord2f(mord[d * HEADS + hd]));
    atomicAdd(&denom[d * HEADS + hd], ex);
}

// ---------------------------------------------------------------------------
// 5) scatter: acc[dst,hd,:] += alpha * h[src,hd,:]
//    one lane per channel -> 32-lane slice per (edge, head); fully coalesced
//    128 B gathers and coalesced global_atomic_add_f32 into L2-resident acc.
// ---------------------------------------------------------------------------
__global__ void aggregate_kernel(const long long* __restrict__ ei,
                                 const float* __restrict__ h,
                                 const float* __restrict__ as, const float* __restrict__ ad,
                                 const unsigned* __restrict__ mord,
                                 const float* __restrict__ denom,
                                 float* __restrict__ acc, int E_, int ET) {
    int t = blockIdx.x * blockDim.x + threadIdx.x;   // ET*128 threads
    if (t >= ET * OUTC) return;
    int c   = t & 31;
    int hd  = (t >> 5) & 3;
    int idx = t >> 7;
    int s, d; edge_sd(ei, idx, E_, s, d);
    float e = as[s * HEADS + hd] + ad[d * HEADS + hd];
    e = (e >= 0.0f) ? e : NEG_SLOPE * e;
    float ex = expf(e - ord2f(mord[d * HEADS + hd]));
    float alpha = ex / (denom[d * HEADS + hd] + 1e-16f);
    float val = alpha * h[(size_t)s * OUTC + hd * CPH + c];
    atomicAdd(&acc[(size_t)d * OUTC + hd * CPH + c], val);
}

// ---------------------------------------------------------------------------
// 6) batch-norm statistics: channel = lane, block strides over nodes
// ---------------------------------------------------------------------------
__global__ void stats_kernel(const float* __restrict__ acc, const float* __restrict__ bias,
                             float* __restrict__ sums, float* __restrict__ sumsq, int n) {
    int ch = threadIdx.x;            // blockDim.x == 128
    float b = bias[ch];
    float s = 0.0f, sq = 0.0f;
    for (int node = blockIdx.x; node < n; node += gridDim.x) {
        float v = acc[(size_t)node * OUTC + ch] + b;
        s += v; sq += v * v;
    }
    atomicAdd(&sums[ch], s);
    atomicAdd(&sumsq[ch], sq);
}

// ---------------------------------------------------------------------------
// 7) normalize + affine + ELU
// ---------------------------------------------------------------------------
__global__ void finalize_kernel(const float* __restrict__ acc, const float* __restrict__ bias,
                                const float* __restrict__ sums, const float* __restrict__ sumsq,
                                const float* __restrict__ gamma, const float* __restrict__ beta,
                                float* __restrict__ out, int n) {
    int t = blockIdx.x * blockDim.x + threadIdx.x;
    if (t >= n * OUTC) return;
    int ch = t & (OUTC - 1);
    float invN = 1.0f / (float)n;
    float mean = sums[ch] * invN;
    float var  = sumsq[ch] * invN - mean * mean;
    float v = acc[t] + bias[ch];
    v = (v - mean) * rsqrtf(var + BN_EPS) * gamma[ch] + beta[ch];
    out[t] = (v > 0.0f) ? v : (expf(v) - 1.0f);
}

// ---------------------------------------------------------------------------
extern "C" void kernel_launch(void* const* d_in, const int* in_sizes, int n_in,
                              void* d_out, int out_size, void* d_ws, size_t ws_size,
                              hipStream_t stream) {
    const float*     x       = (const float*)d_in[0];
    const long long* ei      = (const long long*)d_in[1];   // int64 edge_index [2,E]
    const float*     W       = (const float*)d_in[2];
    const float*     att_src = (const float*)d_in[3];
    const float*     att_dst = (const float*)d_in[4];
    const float*     bias    = (const float*)d_in[5];
    const float*     gamma   = (const float*)d_in[6];
    const float*     beta    = (const float*)d_in[7];
    float*           out     = (float*)d_out;

    const int n  = in_sizes[0] / INC;   // 50000
    const int E_ = in_sizes[1] / 2;     // 800000
    const int ET = E_ + n;              // + self loops

    // workspace carve-out
    char* ws = (char*)d_ws;
    float*    h     = (float*)ws;    ws += (size_t)n * OUTC * sizeof(float);
    float*    acc   = (float*)ws;    ws += (size_t)n * OUTC * sizeof(float);
    float*    as    = (float*)ws;    ws += (size_t)n * HEADS * sizeof(float);
    float*    ad    = (float*)ws;    ws += (size_t)n * HEADS * sizeof(float);
    unsigned* mord  = (unsigned*)ws; ws += (size_t)n * HEADS * sizeof(unsigned);
    float*    denom = (float*)ws;    ws += (size_t)n * HEADS * sizeof(float);
    float*    sums  = (float*)ws;    ws += OUTC * sizeof(float);
    float*    sumsq = (float*)ws;

    init_ws_kernel<<<2048, 256, 0, stream>>>(acc, mord, denom, sums, sumsq, n);

    // GEMM: one block per 16-row stripe (N = 50000 is a multiple of 16)
    gemm_xw_wmma<<<n / 16, 256, 0, stream>>>(x, W, h);

    attn_coef_kernel<<<(n * HEADS + 255) / 256, 256, 0, stream>>>(h, att_src, att_dst, as, ad, n);

    int eh_threads = ET * HEADS;
    edge_max_kernel<<<(eh_threads + 255) / 256, 256, 0, stream>>>(ei, as, ad, mord, E_, ET);
    edge_sum_kernel<<<(eh_threads + 255) / 256, 256, 0, stream>>>(ei, as, ad, mord, denom, E_, ET);

    int agg_threads = ET * OUTC;   // 108.8M, fits in int32
    aggregate_kernel<<<(agg_threads + 255) / 256, 256, 0, stream>>>(ei, h, as, ad, mord, denom,
                                                                    acc, E_, ET);

    stats_kernel<<<512, 128, 0, stream>>>(acc, bias, sums, sumsq, n);

    finalize_kernel<<<(n * OUTC + 255) / 256, 256, 0, stream>>>(acc, bias, sums, sumsq,
                                                                gamma, beta, out, n);
}